// SpatioRegionSelection_52553219833981
// MI455X (gfx1250) — compile-verified
//
#include <hip/hip_runtime.h>
#include <hip/hip_bf16.h>

// Problem constants
#define B_   16
#define T_   60
#define N_   196
#define C_   512
#define SEGS 10
#define TOPK 6
#define TOPM 12
#define HEADS 4
#define HD   128
#define CLIP 6           // T/SEGS
#define FRAMES 36        // CLIP*TOPK

typedef __attribute__((ext_vector_type(2))) float v2f;
typedef __attribute__((ext_vector_type(8))) float v8f;

// ---------------------------------------------------------------------------
// K1: q[c] = qst_feat[0,:] . Wq[c,:] + bq[c]   (Wq = in_proj_w rows [0,512))
// grid 64 x 256 : 8 rows per block, one wave (32 lanes) per row.
// ---------------------------------------------------------------------------
__global__ void __launch_bounds__(256) srs_q_kernel(
    const float* __restrict__ qst, const float* __restrict__ W,
    const float* __restrict__ bias, float* __restrict__ q_ws) {
  int lane = threadIdx.x & 31;
  int row  = threadIdx.x >> 5;                 // 0..7
  int c    = blockIdx.x * 8 + row;             // 0..511
  const float* wr = W + (size_t)c * C_;
  float p = 0.f;
  #pragma unroll 4
  for (int s = lane; s < C_; s += 32) p += qst[s] * wr[s];
  #pragma unroll
  for (int off = 16; off; off >>= 1) p += __shfl_down(p, off, 32);
  if (lane == 0) q_ws[c] = p + bias[c];
}

// ---------------------------------------------------------------------------
// K2: r[c'][h] = sum_d q[h*128+d] * Wk[(512+h*128+d), c']
// (bk folds to a per-head additive constant -> cancels in softmax, dropped)
// grid 8 x 256 : block g -> head g>>1, c' half g&1. Coalesced column walks.
// ---------------------------------------------------------------------------
__global__ void __launch_bounds__(256) srs_r_kernel(
    const float* __restrict__ W, const float* __restrict__ q_ws,
    float* __restrict__ r_ws) {
  __shared__ float qs[HD];
  int t = threadIdx.x, g = blockIdx.x;
  int h  = g >> 1;
  int cp = ((g & 1) << 8) + t;                 // 0..511
  if (t < HD) qs[t] = q_ws[h * HD + t];
  __syncthreads();
  const float* wb = W + ((size_t)(C_ + h * HD)) * C_ + cp;
  float acc = 0.f;
  #pragma unroll 4
  for (int d = 0; d < HD; ++d) acc += qs[d] * wb[(size_t)d * C_];
  r_ws[cp * 4 + h] = acc;
}

// ---------------------------------------------------------------------------
// K3: scores(196x4) = P(196x512) @ r^T(512x4) * 1/sqrt(HD) via
//     V_WMMA_F32_16X16X4_F32, then per-head softmax over N, head-mean,
//     stable top-12 membership, pid = max index in top set.
// One workgroup, 8 waves, 13 M-tiles of 16 rows.
//
// Divergence-free inner loop:
//  * A rows 196..207 (last tile) are loaded UNGUARDED: f0 <= 54 so those
//    addresses land in frame f0+1 of batch 0 (in-bounds); the garbage only
//    reaches D rows that are masked out at the LDS score write (m < 196).
//  * B operand comes from a zero-padded [512][16] LDS tile so cols 4..15
//    read zeros with no predication. EXEC stays all-1s around every WMMA.
// ---------------------------------------------------------------------------
__global__ void __launch_bounds__(256) srs_pid_kernel(
    const float* __restrict__ patch, const int* __restrict__ idx,
    const float* __restrict__ r_ws, int* __restrict__ pid_ws) {
  __shared__ float rb[C_ * 16];                // B tile: [k][16], cols>=4 zero
  __shared__ float sc[N_ * 4];                 // scores [n*4 + h]
  __shared__ float pw[N_];
  __shared__ float hmax[HEADS], hsum[HEADS];
  __shared__ int   pid_sh;

  int t = threadIdx.x;
  for (int i = t; i < C_ * 16; i += 256) {
    int k = i >> 4, col = i & 15;
    rb[i] = (col < HEADS) ? r_ws[k * 4 + col] : 0.f;
  }
  if (t == 0) pid_sh = 0;
  __syncthreads();

  int lane = t & 31, wave = t >> 5;
  int f0 = idx[0] * CLIP;                      // batch 0, frame 0 (<= 54)
  const float* P = patch + (size_t)f0 * N_ * C_;   // b=0 plane
  const float scale = 0.08838834764831845f;    // 1/sqrt(128)

  int hi  = lane >> 4;                         // half-wave select
  int col = lane & 15;                         // N index (head if < 4)

  for (int mt = wave; mt < 13; mt += 8) {      // wave-uniform loop
    int m0  = mt * 16;
    int row = m0 + (lane & 15);
    const float* prow = P + (size_t)row * C_;  // always in-bounds (see above)
    const float* bcol0 = &rb[((hi << 1)) * 16 + col];
    v8f acc = {0.f, 0.f, 0.f, 0.f, 0.f, 0.f, 0.f, 0.f};
    for (int k0 = 0; k0 < C_; k0 += 4) {
      int kr = k0 + (hi << 1);                 // K pair handled by this half
      v2f a, bv;
      a.x  = prow[kr];
      a.y  = prow[kr + 1];
      bv.x = bcol0[(size_t)k0 * 16];           // rb[kr][col]
      bv.y = bcol0[(size_t)k0 * 16 + 16];      // rb[kr+1][col]
      acc = __builtin_amdgcn_wmma_f32_16x16x4_f32(
          /*neg_a=*/false, a, /*neg_b=*/false, bv,
          /*c_mod=*/(short)0, acc, /*reuse_a=*/false, /*reuse_b=*/false);
    }
    // D layout: VGPR v -> M = m0 + v (+8 for upper half), N = col
    int mbase = m0 + (hi << 3);
    if (col < HEADS) {
      #pragma unroll
      for (int v = 0; v < 8; ++v) {
        int m = mbase + v;
        if (m < N_) sc[m * 4 + col] = acc[v] * scale;
      }
    }
  }
  __syncthreads();

  if (t < HEADS) {                             // per-head softmax stats
    float mx = -1e30f;
    for (int m = 0; m < N_; ++m) mx = fmaxf(mx, sc[m * 4 + t]);
    float sm = 0.f;
    for (int m = 0; m < N_; ++m) sm += expf(sc[m * 4 + t] - mx);
    hmax[t] = mx; hsum[t] = sm;
  }
  __syncthreads();

  if (t < N_) {                                // head-mean attention weight
    float s = 0.f;
    #pragma unroll
    for (int h = 0; h < HEADS; ++h) s += expf(sc[t * 4 + h] - hmax[h]) / hsum[h];
    pw[t] = 0.25f * s;
  }
  __syncthreads();

  if (t < N_) {                                // stable-argsort top-12 rule
    float mine = pw[t];
    int cnt = 0;
    for (int m = 0; m < N_; ++m) {
      float o = pw[m];
      cnt += (o > mine) || (o == mine && m > t);  // ties: larger idx ranks higher
    }
    if (cnt < TOPM) atomicMax(&pid_sh, t);     // pid = max index in top set
  }
  __syncthreads();
  if (t == 0) pid_ws[0] = pid_sh;
}

// ---------------------------------------------------------------------------
// K4: outputs. One block per (b,f):
//   out0: audio gather (128 float4)
//   out1/out2: broadcast patch_feat[b,fid,pid,:] 12x into each region.
// 29.5 MB of coalesced float4 stores -> HBM-write bound (~1.3 us ideal).
// ---------------------------------------------------------------------------
__global__ void __launch_bounds__(256) srs_out_kernel(
    const float* __restrict__ audio, const float* __restrict__ patch,
    const int* __restrict__ idx, const int* __restrict__ pid_ws,
    float* __restrict__ out) {
  int g = blockIdx.x, t = threadIdx.x;
  int b = g / FRAMES, f = g % FRAMES;
  int fid = idx[b * TOPK + f / CLIP] * CLIP + (f % CLIP);
  int pid = pid_ws[0];
  pid = pid < 0 ? 0 : (pid > N_ - 1 ? N_ - 1 : pid);

  float4* out4 = (float4*)out;
  const int q = t >> 7;            // 0/1 : which half of the 12 copies
  const int e = t & 127;           // float4 index within 512-float row

  // audio_top_k
  if (t < 128) {
    const float4* asrc = (const float4*)(audio + ((size_t)b * T_ + fid) * C_);
    out4[(size_t)g * 128 + t] = asrc[t];
  }

  // selected patch row (cached broadcast read)
  const float4* psrc =
      (const float4*)(patch + (((size_t)b * T_ + fid) * N_ + pid) * C_);
  float4 v = psrc[e];

  const size_t base1 = (size_t)B_ * FRAMES * C_ / 4;                 // 73728
  const size_t base2 = base1 + (size_t)B_ * FRAMES * TOPM * C_ / 4;  // 958464
  #pragma unroll
  for (int j = 0; j < 6; ++j) {
    int m = q * 6 + j;
    size_t off = ((size_t)g * TOPM + m) * 128 + e;
    out4[base1 + off] = v;   // visual_patch_top_m
    out4[base2 + off] = v;   // visual_patch_feat (identical data)
  }
}

// ---------------------------------------------------------------------------
extern "C" void kernel_launch(void* const* d_in, const int* in_sizes, int n_in,
                              void* d_out, int out_size, void* d_ws, size_t ws_size,
                              hipStream_t stream) {
  const float* audio = (const float*)d_in[0];   // (B,T,C)
  const float* patch = (const float*)d_in[1];   // (B,T,N,C)
  const float* qst   = (const float*)d_in[2];   // (B,C)
  const int*   idx   = (const int*)d_in[3];     // (B,1,TOPK) int32
  const float* W     = (const float*)d_in[4];   // (3C,C)
  const float* bias  = (const float*)d_in[5];   // (3C,)
  float* out = (float*)d_out;

  float* wsf   = (float*)d_ws;
  float* q_ws  = wsf;            // 512 floats
  float* r_ws  = wsf + 512;      // 2048 floats (interleaved [c*4+h])
  int*   pid_w = (int*)(wsf + 2560);

  srs_q_kernel  <<<64, 256, 0, stream>>>(qst, W, bias, q_ws);
  srs_r_kernel  <<< 8, 256, 0, stream>>>(W, q_ws, r_ws);
  srs_pid_kernel<<< 1, 256, 0, stream>>>(patch, idx, r_ws, pid_w);
  srs_out_kernel<<<B_ * FRAMES, 256, 0, stream>>>(audio, patch, idx, pid_w, out);
}